// CTCGreadySearchDecoder_47132971106739
// MI455X (gfx1250) — compile-verified
//
#include <hip/hip_runtime.h>
#include <math.h>

// ---------------------------------------------------------------------------
// CTC greedy decode for B=8, T=1024, V=10000 (fp32 logits).
// Phase 1: per-row streaming logsumexp + argmax (memory-bound, ~327 MB read).
//          Uses gfx1250 async global->LDS copies (ASYNCcnt) for bulk prefetch.
// Phase 2: per-batch blank/duplicate collapse with an LDS scan.
// ---------------------------------------------------------------------------

#define BATCH       8
#define TLEN        1024
#define VOCAB       10000
#define VEC4        (VOCAB / 4)                     // 2500 float4 per row
#define P1_THREADS  256
#define NSTAGES     ((VEC4 + P1_THREADS - 1) / P1_THREADS)  // 10

#ifndef __has_builtin
#define __has_builtin(x) 0
#endif

#if __has_builtin(__builtin_amdgcn_global_load_async_to_lds_b128)
#define HAVE_ASYNC_LDS 1
#else
#define HAVE_ASYNC_LDS 0
#endif

#if HAVE_ASYNC_LDS
// The builtin's pointer params are to 4 x i32 vectors (per hipcc diagnostic):
//   param0: global (AS1) int4*, param1: LDS (AS3) int4*, then imm offset, imm cpol.
typedef int v4i_t __attribute__((vector_size(4 * sizeof(int))));
typedef __attribute__((address_space(1))) v4i_t gv4i_t;
typedef __attribute__((address_space(3))) v4i_t lv4i_t;

__device__ __forceinline__ void async_copy_b128(const void* g, void* l) {
  __builtin_amdgcn_global_load_async_to_lds_b128(
      (gv4i_t*)(g), (lv4i_t*)(l), /*offset=*/0, /*cpol=*/0);
}

__device__ __forceinline__ void async_wait_all() {
#if __has_builtin(__builtin_amdgcn_s_wait_asynccnt)
  __builtin_amdgcn_s_wait_asynccnt(0);
#else
  asm volatile("s_wait_asynccnt 0" ::: "memory");
#endif
}
#endif // HAVE_ASYNC_LDS

// ---------------------------------------------------------------------------
// Phase 1: one block per (b,t) row. Online logsumexp + first-index argmax.
// ---------------------------------------------------------------------------
__global__ __launch_bounds__(P1_THREADS)
void ctc_row_reduce(const float* __restrict__ logits,
                    int* __restrict__ preds,
                    float* __restrict__ maxlogp)
{
  const int row = blockIdx.x;               // 0 .. B*T-1
  const int tid = threadIdx.x;
  const float4* __restrict__ src =
      reinterpret_cast<const float4*>(logits) + (size_t)row * VEC4;

#if HAVE_ASYNC_LDS
  __shared__ float4 stage[NSTAGES][P1_THREADS];   // 40 KB of the 320 KB WGP pool
  // Issue every stage up-front: 10 outstanding b128 async copies per wave,
  // tracked with ASYNCcnt, no VGPR cost. Tail lanes clamp to a valid address
  // (their data is masked out at consume time).
  #pragma unroll
  for (int k = 0; k < NSTAGES; ++k) {
    int i = k * P1_THREADS + tid;
    if (i >= VEC4) i = VEC4 - 1;
    async_copy_b128(&src[i], &stage[k][tid]);
  }
  async_wait_all();
#endif

  float m = -__builtin_inff();
  float s = 0.0f;
  int   best = 0;

  auto upd = [&](float x, int gi) {
    if (x > m) {                      // strict '>' keeps first occurrence
      s = s * __expf(m - x) + 1.0f;
      m = x;
      best = gi;
    } else {
      s += __expf(x - m);
    }
  };

  #pragma unroll
  for (int k = 0; k < NSTAGES; ++k) {
    const int i = k * P1_THREADS + tid;
#if HAVE_ASYNC_LDS
    float4 v = stage[k][tid];
#else
    float4 v = src[(i < VEC4) ? i : (VEC4 - 1)];
#endif
    if (i < VEC4) {
      const int gi = i * 4;
      upd(v.x, gi + 0);
      upd(v.y, gi + 1);
      upd(v.z, gi + 2);
      upd(v.w, gi + 3);
    }
  }

  // Block-level merge of (m, s, idx) online-softmax states.
  __shared__ float red_m[P1_THREADS];
  __shared__ float red_s[P1_THREADS];
  __shared__ int   red_i[P1_THREADS];
  red_m[tid] = m; red_s[tid] = s; red_i[tid] = best;
  __syncthreads();

  for (int off = P1_THREADS / 2; off > 0; off >>= 1) {
    if (tid < off) {
      float m1 = red_m[tid],       s1 = red_s[tid];       int i1 = red_i[tid];
      float m2 = red_m[tid + off], s2 = red_s[tid + off]; int i2 = red_i[tid + off];
      float mm, ss; int ii;
      if (m1 > m2)      { mm = m1; ss = s1 + s2 * __expf(m2 - m1); ii = i1; }
      else if (m2 > m1) { mm = m2; ss = s2 + s1 * __expf(m1 - m2); ii = i2; }
      else              { mm = m1; ss = s1 + s2; ii = (i1 < i2) ? i1 : i2; }
      red_m[tid] = mm; red_s[tid] = ss; red_i[tid] = ii;
    }
    __syncthreads();
  }

  if (tid == 0) {
    preds[row]   = red_i[0];
    // max(log_softmax) = max(x) - (max(x) + log(sum exp(x - max))) = -log(s)
    maxlogp[row] = -__logf(red_s[0]);
  }
}

// ---------------------------------------------------------------------------
// Phase 2: one block per batch. keep = valid & pred!=BLANK & pred!=prev,
// inclusive scan -> left-pack positions, scatter via LDS (zero-padded),
// score = sum of kept maxlogp.
// ---------------------------------------------------------------------------
__global__ __launch_bounds__(TLEN)
void ctc_collapse(const int* __restrict__ preds,
                  const float* __restrict__ maxlogp,
                  const int* __restrict__ lengths,
                  float* __restrict__ out)
{
  const int b = blockIdx.x;
  const int t = threadIdx.x;

  __shared__ int   scan[TLEN];
  __shared__ int   toks[TLEN];
  __shared__ float sred[TLEN];

  const int pred = preds[b * TLEN + t];
  const int prev = (t == 0) ? 0 : preds[b * TLEN + t - 1];
  const int len  = lengths[b];
  const bool keep = (t < len) && (pred != 0) && (pred != prev);

  scan[t] = keep ? 1 : 0;
  toks[t] = 0;
  sred[t] = keep ? maxlogp[b * TLEN + t] : 0.0f;
  __syncthreads();

  // Hillis-Steele inclusive scan over 1024 elements.
  for (int off = 1; off < TLEN; off <<= 1) {
    int add = (t >= off) ? scan[t - off] : 0;
    __syncthreads();
    scan[t] += add;
    __syncthreads();
  }

  const int total = scan[TLEN - 1];
  if (keep) toks[scan[t] - 1] = pred;   // unique positions, zero-padded base
  __syncthreads();

  out[(size_t)b * TLEN + t] = (float)toks[t];

  // Score sum.
  for (int off = TLEN / 2; off > 0; off >>= 1) {
    if (t < off) sred[t] += sred[t + off];
    __syncthreads();
  }

  if (t == 0) {
    out[BATCH * TLEN + b]         = (float)total;   // out_len
    out[BATCH * TLEN + BATCH + b] = sred[0];        // scores
  }
}

// ---------------------------------------------------------------------------
extern "C" void kernel_launch(void* const* d_in, const int* in_sizes, int n_in,
                              void* d_out, int out_size, void* d_ws, size_t ws_size,
                              hipStream_t stream) {
  const float* logits  = (const float*)d_in[0];   // [B, T, V] fp32
  const int*   lengths = (const int*)d_in[1];     // [B] int32
  float*       out     = (float*)d_out;           // [B*T tokens | B out_len | B scores]

  int*   preds = (int*)d_ws;                                          // B*T int32
  float* mlp   = (float*)((char*)d_ws + sizeof(int) * BATCH * TLEN);  // B*T fp32

  ctc_row_reduce<<<BATCH * TLEN, P1_THREADS, 0, stream>>>(logits, preds, mlp);
  ctc_collapse<<<BATCH, TLEN, 0, stream>>>(preds, mlp, lengths, out);
}